// Net_58712202936903
// MI455X (gfx1250) — compile-verified
//
#include <hip/hip_runtime.h>

// ---------------------------------------------------------------------------
// CDNA5 (gfx1250) fused 2-layer multiplicative MLP + JVP, fp32 WMMA path.
//
// y1 = (x W1_1^T + b1_1) * (x W2_1^T + b2_1)          (8 -> 128)
// y2 = (y1 W1_2^T + b1_2) * (y1 W2_2^T + b2_2)        (128 -> 128)
// out = y2 . w_out                                    (128 -> 1)
// grad term == JVP: propagate tangent ydot with product rule, dot with w_out.
//
// One wave32 handles one 16-sample tile using V_WMMA_F32_16X16X4_F32.
// ---------------------------------------------------------------------------

typedef float v2f __attribute__((ext_vector_type(2)));
typedef float v8f __attribute__((ext_vector_type(8)));

#define H 128
#define NIN 8
#define LDS_STRIDE 130  // 128 + 2 pad -> conflict-free ds_load_b64 A-fragments

__device__ __forceinline__ v8f wmma_k4(v2f a, v2f b, v8f c) {
  // V_WMMA_F32_16X16X4_F32 : D = A(16x4) x B(4x16) + C(16x16), all f32
  return __builtin_amdgcn_wmma_f32_16x16x4_f32(
      /*neg_a=*/false, a, /*neg_b=*/false, b,
      /*c_mod=*/(short)0, c, /*reuse_a=*/false, /*reuse_b=*/false);
}

__device__ __forceinline__ v2f ldg2(const float* p) {
  return *reinterpret_cast<const v2f*>(p);
}

// ------------------------------ forward only -------------------------------
template <int WAVES>
__global__ __launch_bounds__(WAVES * 32) void net_forward_kernel(
    const float* __restrict__ x, float* __restrict__ out, int n_samples,
    const float* __restrict__ w1_1, const float* __restrict__ b1_1,
    const float* __restrict__ w2_1, const float* __restrict__ b2_1,
    const float* __restrict__ w1_2, const float* __restrict__ b1_2,
    const float* __restrict__ w2_2, const float* __restrict__ b2_2,
    const float* __restrict__ w_out) {
  __shared__ float ybuf[WAVES][16 * LDS_STRIDE];

  const int wave = threadIdx.x >> 5;
  const int lane = threadIdx.x & 31;
  const int half = lane >> 4;   // A/B K-split, C M-split at lane 16
  const int l16 = lane & 15;
  const int tile = blockIdx.x * WAVES + wave;
  if (tile * 16 >= n_samples) return;  // wave-uniform: EXEC stays all-ones
  const int row0 = tile * 16;
  float* yb = ybuf[wave];

  // ---- layer 1 (K = 8 -> two K4 steps), write y1 tile to LDS
  {
    const float* xr = x + (row0 + l16) * NIN + 2 * half;
    v2f a0 = ldg2(xr);       // K = 0..3
    v2f a1 = ldg2(xr + 4);   // K = 4..7
#pragma unroll
    for (int nt = 0; nt < 8; ++nt) {
      const int n = nt * 16 + l16;
      v2f p10 = ldg2(w1_1 + n * NIN + 2 * half);
      v2f p11 = ldg2(w1_1 + n * NIN + 4 + 2 * half);
      v2f p20 = ldg2(w2_1 + n * NIN + 2 * half);
      v2f p21 = ldg2(w2_1 + n * NIN + 4 + 2 * half);
      v8f c1 = {}, c2 = {};
      c1 = wmma_k4(a0, p10, c1);
      c1 = wmma_k4(a1, p11, c1);
      c2 = wmma_k4(a0, p20, c2);
      c2 = wmma_k4(a1, p21, c2);
      const float bb1 = b1_1[n], bb2 = b2_1[n];
#pragma unroll
      for (int j = 0; j < 8; ++j) {
        const int m = j + 8 * half;
        yb[m * LDS_STRIDE + n] = (c1[j] + bb1) * (c2[j] + bb2);
      }
    }
  }

  // ---- layer 2 (K = 128) fused with output dot
  float part[8];
#pragma unroll
  for (int j = 0; j < 8; ++j) part[j] = 0.f;

  const float* arow = yb + l16 * LDS_STRIDE + 2 * half;
#pragma unroll
  for (int nt = 0; nt < 8; ++nt) {
    const int n = nt * 16 + l16;
    const float* b1r = w1_2 + n * H + 2 * half;
    const float* b2r = w2_2 + n * H + 2 * half;
    v8f c1 = {}, c2 = {};
#pragma unroll 4
    for (int ks = 0; ks < 32; ++ks) {
      v2f a = *reinterpret_cast<const v2f*>(arow + ks * 4);  // ds_load_b64
      v2f q1 = ldg2(b1r + ks * 4);
      v2f q2 = ldg2(b2r + ks * 4);
      c1 = wmma_k4(a, q1, c1);
      c2 = wmma_k4(a, q2, c2);
    }
    const float bia1 = b1_2[n], bia2 = b2_2[n], wo = w_out[n];
#pragma unroll
    for (int j = 0; j < 8; ++j)
      part[j] += (c1[j] + bia1) * (c2[j] + bia2) * wo;
  }

  // butterfly over 16-lane halves (wave32)
#pragma unroll
  for (int j = 0; j < 8; ++j) {
    float v = part[j];
    v += __shfl_xor(v, 1, 32);
    v += __shfl_xor(v, 2, 32);
    v += __shfl_xor(v, 4, 32);
    v += __shfl_xor(v, 8, 32);
    part[j] = v;
  }
  if (l16 == 0) {
#pragma unroll
    for (int j = 0; j < 8; ++j) out[row0 + j + 8 * half] = part[j];
  }
}

// --------------------------- forward + JVP (grad) --------------------------
template <int WAVES>
__global__ __launch_bounds__(WAVES * 32) void net_fwd_grad_kernel(
    const float* __restrict__ x, const float* __restrict__ xdot,
    float* __restrict__ outY, float* __restrict__ outG, int n_samples,
    const float* __restrict__ w1_1, const float* __restrict__ b1_1,
    const float* __restrict__ w2_1, const float* __restrict__ b2_1,
    const float* __restrict__ w1_2, const float* __restrict__ b1_2,
    const float* __restrict__ w2_2, const float* __restrict__ b2_2,
    const float* __restrict__ w_out) {
  __shared__ float ybuf[WAVES][16 * LDS_STRIDE];
  __shared__ float ydbuf[WAVES][16 * LDS_STRIDE];

  const int wave = threadIdx.x >> 5;
  const int lane = threadIdx.x & 31;
  const int half = lane >> 4;
  const int l16 = lane & 15;
  const int tile = blockIdx.x * WAVES + wave;
  if (tile * 16 >= n_samples) return;
  const int row0 = tile * 16;
  float* yb = ybuf[wave];
  float* ydb = ydbuf[wave];

  // ---- layer 1 value + tangent
  {
    const float* xr = x + (row0 + l16) * NIN + 2 * half;
    const float* xdr = xdot + (row0 + l16) * NIN + 2 * half;
    v2f a0 = ldg2(xr), a1 = ldg2(xr + 4);
    v2f d0 = ldg2(xdr), d1 = ldg2(xdr + 4);
#pragma unroll
    for (int nt = 0; nt < 8; ++nt) {
      const int n = nt * 16 + l16;
      v2f p10 = ldg2(w1_1 + n * NIN + 2 * half);
      v2f p11 = ldg2(w1_1 + n * NIN + 4 + 2 * half);
      v2f p20 = ldg2(w2_1 + n * NIN + 2 * half);
      v2f p21 = ldg2(w2_1 + n * NIN + 4 + 2 * half);
      v8f c1 = {}, c2 = {}, g1 = {}, g2 = {};
      c1 = wmma_k4(a0, p10, c1);
      c1 = wmma_k4(a1, p11, c1);
      c2 = wmma_k4(a0, p20, c2);
      c2 = wmma_k4(a1, p21, c2);
      g1 = wmma_k4(d0, p10, g1);
      g1 = wmma_k4(d1, p11, g1);
      g2 = wmma_k4(d0, p20, g2);
      g2 = wmma_k4(d1, p21, g2);
      const float bb1 = b1_1[n], bb2 = b2_1[n];
#pragma unroll
      for (int j = 0; j < 8; ++j) {
        const int m = j + 8 * half;
        const float z1 = c1[j] + bb1, z2 = c2[j] + bb2;
        yb[m * LDS_STRIDE + n] = z1 * z2;
        ydb[m * LDS_STRIDE + n] = g1[j] * z2 + z1 * g2[j];  // product rule
      }
    }
  }

  // ---- layer 2 value + tangent, fused output dots
  float pY[8], pG[8];
#pragma unroll
  for (int j = 0; j < 8; ++j) {
    pY[j] = 0.f;
    pG[j] = 0.f;
  }

  const float* arow = yb + l16 * LDS_STRIDE + 2 * half;
  const float* drow = ydb + l16 * LDS_STRIDE + 2 * half;
#pragma unroll
  for (int nt = 0; nt < 8; ++nt) {
    const int n = nt * 16 + l16;
    const float* b1r = w1_2 + n * H + 2 * half;
    const float* b2r = w2_2 + n * H + 2 * half;
    v8f c1 = {}, c2 = {}, g1 = {}, g2 = {};
#pragma unroll 4
    for (int ks = 0; ks < 32; ++ks) {
      v2f a = *reinterpret_cast<const v2f*>(arow + ks * 4);
      v2f ad = *reinterpret_cast<const v2f*>(drow + ks * 4);
      v2f q1 = ldg2(b1r + ks * 4);
      v2f q2 = ldg2(b2r + ks * 4);
      c1 = wmma_k4(a, q1, c1);
      c2 = wmma_k4(a, q2, c2);
      g1 = wmma_k4(ad, q1, g1);
      g2 = wmma_k4(ad, q2, g2);
    }
    const float bia1 = b1_2[n], bia2 = b2_2[n], wo = w_out[n];
#pragma unroll
    for (int j = 0; j < 8; ++j) {
      const float z1 = c1[j] + bia1, z2 = c2[j] + bia2;
      pY[j] += z1 * z2 * wo;
      pG[j] += (g1[j] * z2 + z1 * g2[j]) * wo;
    }
  }

#pragma unroll
  for (int j = 0; j < 8; ++j) {
    float vy = pY[j], vg = pG[j];
    vy += __shfl_xor(vy, 1, 32);
    vy += __shfl_xor(vy, 2, 32);
    vy += __shfl_xor(vy, 4, 32);
    vy += __shfl_xor(vy, 8, 32);
    vg += __shfl_xor(vg, 1, 32);
    vg += __shfl_xor(vg, 2, 32);
    vg += __shfl_xor(vg, 4, 32);
    vg += __shfl_xor(vg, 8, 32);
    pY[j] = vy;
    pG[j] = vg;
  }
  if (l16 == 0) {
#pragma unroll
    for (int j = 0; j < 8; ++j) {
      outY[row0 + j + 8 * half] = pY[j];
      outG[row0 + j + 8 * half] = pG[j];
    }
  }
}

// ------------------------------- launcher ----------------------------------
extern "C" void kernel_launch(void* const* d_in, const int* in_sizes, int n_in,
                              void* d_out, int out_size, void* d_ws,
                              size_t ws_size, hipStream_t stream) {
  const float* T = (const float*)d_in[0];
  const float* l = (const float*)d_in[1];
  const float* l1_dot = (const float*)d_in[2];
  const float* center = (const float*)d_in[3];
  const float* w1_1 = (const float*)d_in[4];
  const float* b1_1 = (const float*)d_in[5];
  const float* w2_1 = (const float*)d_in[6];
  const float* b2_1 = (const float*)d_in[7];
  const float* w1_2 = (const float*)d_in[8];
  const float* b1_2 = (const float*)d_in[9];
  const float* w2_2 = (const float*)d_in[10];
  const float* b2_2 = (const float*)d_in[11];
  const float* w_out = (const float*)d_in[12];

  const int nT = in_sizes[0] / NIN;  // 65536
  const int nL = in_sizes[1] / NIN;  // 65536
  const int nC = in_sizes[3] / NIN;  // 512

  float* out = (float*)d_out;
  float* v_t = out;
  float* v_y = out + nT;
  float* v_g = out + nT + nL;
  float* v_c = out + nT + 2 * nL;

  constexpr int WF = 4;  // waves/block, forward kernels (33 KB LDS)
  constexpr int WG = 2;  // waves/block, grad kernel    (33 KB LDS)

  const int tilesT = (nT + 15) / 16;
  const int tilesL = (nL + 15) / 16;
  const int tilesC = (nC + 15) / 16;

  net_forward_kernel<WF><<<(tilesT + WF - 1) / WF, WF * 32, 0, stream>>>(
      T, v_t, nT, w1_1, b1_1, w2_1, b2_1, w1_2, b1_2, w2_2, b2_2, w_out);

  net_forward_kernel<WF><<<(tilesC + WF - 1) / WF, WF * 32, 0, stream>>>(
      center, v_c, nC, w1_1, b1_1, w2_1, b2_1, w1_2, b1_2, w2_2, b2_2, w_out);

  net_fwd_grad_kernel<WG><<<(tilesL + WG - 1) / WG, WG * 32, 0, stream>>>(
      l, l1_dot, v_y, v_g, nL, w1_1, b1_1, w2_1, b2_1, w1_2, b1_2, w2_2, b2_2,
      w_out);
}